// MultiLayerRGN_38319698215246
// MI455X (gfx1250) — compile-verified
//
#include <hip/hip_runtime.h>
#include <hip/hip_bf16.h>

typedef float v2f __attribute__((ext_vector_type(2)));
typedef float v8f __attribute__((ext_vector_type(8)));

// ---------------------------------------------------------------------------
// Zero-fill kernel (agg must be zeroed before each edge-scatter pass).
// ---------------------------------------------------------------------------
__global__ __launch_bounds__(256) void zero_f32(float* __restrict__ p, long n) {
    long i = (long)blockIdx.x * blockDim.x + threadIdx.x;
    long stride = (long)gridDim.x * blockDim.x;
    for (; i < n; i += stride) p[i] = 0.0f;
}

// ---------------------------------------------------------------------------
// Edge message kernel, 64 edges per workgroup (4 x 16-row WMMA subtiles).
//   Stage z = [h[src] | ef | h[dst]] for 64 edges into LDS, then each of 8
//   waves owns one 16-column slab of the 128-wide output.  Per K-step the
//   wave loads ONE B fragment from the weight matrix and feeds it to FOUR
//   v_wmma_f32_16x16x4_f32 (one per row subtile), quadrupling B reuse.
//   Epilogue: bias + ReLU, mask by etype, fp32 atomics into agg[dst].
// ---------------------------------------------------------------------------
template <int DIN>
__global__ __launch_bounds__(256) void edge_msg_kernel(
    const float* __restrict__ h,      // [N, DIN]
    const float* __restrict__ ef,     // [E, 32]
    const int*   __restrict__ src,
    const int*   __restrict__ dst,
    const int*   __restrict__ etype,
    const float* __restrict__ mw,     // [3, MSG, 128]
    const float* __restrict__ mb,     // [3, 128]
    float*       __restrict__ agg,    // [N, 384]
    int E)
{
    constexpr int MSG  = 2 * DIN + 32;
    constexpr int MT   = 64;                      // edge rows per workgroup
    constexpr int LDSW = MSG + 4;                 // pad to dodge bank conflicts
    __shared__ float zt[MT * LDSW];

    const int e0 = blockIdx.x * MT;

    // Cooperative gather of the 64 x MSG z-tile into LDS.
    for (int idx = threadIdx.x; idx < MT * MSG; idx += 256) {
        int r = idx / MSG, c = idx % MSG;
        int e = e0 + r;
        float v = 0.0f;
        if (e < E) {
            if (c < DIN)            v = h[(size_t)src[e] * DIN + c];
            else if (c < DIN + 32)  v = ef[(size_t)e * 32 + (c - DIN)];
            else                    v = h[(size_t)dst[e] * DIN + (c - DIN - 32)];
        }
        zt[r * LDSW + c] = v;
    }
    __syncthreads();

    const int wave = threadIdx.x >> 5;
    const int lane = threadIdx.x & 31;
    const int nloc = lane & 15;        // N (column) within slab; also M row within subtile
    const int half = lane >> 4;        // upper/lower half-wave
    const int col  = wave * 16 + nloc; // output column 0..127

    for (int t = 0; t < 3; ++t) {
        const float* __restrict__ mwt = mw + (size_t)t * MSG * 128;
        v8f acc0 = {}, acc1 = {}, acc2 = {}, acc3 = {};
        for (int k0 = 0; k0 < MSG; k0 += 4) {
            const int ka = k0 + 2 * half;
            v2f b;
            b.x = mwt[(size_t)ka * 128 + col];
            b.y = mwt[(size_t)(ka + 1) * 128 + col];
            if (k0 + 8 <= MSG)
                __builtin_prefetch(mwt + (size_t)(k0 + 4) * 128 + col, 0, 3);
            v2f a0, a1, a2, a3;
            a0.x = zt[(nloc)      * LDSW + ka];  a0.y = zt[(nloc)      * LDSW + ka + 1];
            a1.x = zt[(16 + nloc) * LDSW + ka];  a1.y = zt[(16 + nloc) * LDSW + ka + 1];
            a2.x = zt[(32 + nloc) * LDSW + ka];  a2.y = zt[(32 + nloc) * LDSW + ka + 1];
            a3.x = zt[(48 + nloc) * LDSW + ka];  a3.y = zt[(48 + nloc) * LDSW + ka + 1];
            acc0 = __builtin_amdgcn_wmma_f32_16x16x4_f32(false, a0, false, b, (short)0, acc0, false, false);
            acc1 = __builtin_amdgcn_wmma_f32_16x16x4_f32(false, a1, false, b, (short)0, acc1, false, false);
            acc2 = __builtin_amdgcn_wmma_f32_16x16x4_f32(false, a2, false, b, (short)0, acc2, false, false);
            acc3 = __builtin_amdgcn_wmma_f32_16x16x4_f32(false, a3, false, b, (short)0, acc3, false, false);
        }
        const float bias = mb[t * 128 + col];
        v8f accs[4] = {acc0, acc1, acc2, acc3};
#pragma unroll
        for (int i = 0; i < 4; ++i) {
#pragma unroll
            for (int v = 0; v < 8; ++v) {
                const int m = 16 * i + v + 8 * half;   // edge row within tile
                const int e = e0 + m;
                if (e < E && etype[e] == t) {
                    float val = accs[i][v] + bias;
                    val = val > 0.0f ? val : 0.0f;
                    atomicAdd(&agg[(size_t)dst[e] * 384 + t * 128 + col], val);
                }
            }
        }
    }
}

// ---------------------------------------------------------------------------
// Node update kernel: u = concat(h, agg, [nf0]) @ nw + nb, optional ReLU.
// 32 nodes per workgroup (2 x 16-row subtiles), 8 waves cover 128 columns;
// each B fragment feeds two WMMAs.
// ---------------------------------------------------------------------------
template <int DIN, bool CONCAT_NF, bool ACT>
__global__ __launch_bounds__(256) void node_update_kernel(
    const float* __restrict__ h,      // [N, DIN]
    const float* __restrict__ agg,    // [N, 384]
    const float* __restrict__ nf0,    // [N, 64]
    const float* __restrict__ nw,     // [K, 128]
    const float* __restrict__ nb,     // [128]
    float*       __restrict__ out,    // [N, 128]
    int N)
{
    constexpr int K    = DIN + 384 + (CONCAT_NF ? 64 : 0);
    constexpr int MT   = 32;
    constexpr int LDSW = K + 4;
    __shared__ float at[MT * LDSW];

    const int n0 = blockIdx.x * MT;

    for (int idx = threadIdx.x; idx < MT * K; idx += 256) {
        int r = idx / K, c = idx % K;
        int node = n0 + r;
        float v = 0.0f;
        if (node < N) {
            if (c < DIN)            v = h[(size_t)node * DIN + c];
            else if (c < DIN + 384) v = agg[(size_t)node * 384 + (c - DIN)];
            else                    v = nf0[(size_t)node * 64 + (c - DIN - 384)];
        }
        at[r * LDSW + c] = v;
    }
    __syncthreads();

    const int wave = threadIdx.x >> 5;
    const int lane = threadIdx.x & 31;
    const int nloc = lane & 15;
    const int half = lane >> 4;
    const int col  = wave * 16 + nloc;

    v8f acc0 = {}, acc1 = {};
    for (int k0 = 0; k0 < K; k0 += 4) {
        const int ka = k0 + 2 * half;
        v2f b;
        b.x = nw[(size_t)ka * 128 + col];
        b.y = nw[(size_t)(ka + 1) * 128 + col];
        if (k0 + 8 <= K)
            __builtin_prefetch(nw + (size_t)(k0 + 4) * 128 + col, 0, 3);
        v2f a0, a1;
        a0.x = at[(nloc)      * LDSW + ka];  a0.y = at[(nloc)      * LDSW + ka + 1];
        a1.x = at[(16 + nloc) * LDSW + ka];  a1.y = at[(16 + nloc) * LDSW + ka + 1];
        acc0 = __builtin_amdgcn_wmma_f32_16x16x4_f32(false, a0, false, b, (short)0, acc0, false, false);
        acc1 = __builtin_amdgcn_wmma_f32_16x16x4_f32(false, a1, false, b, (short)0, acc1, false, false);
    }
    const float bias = nb[col];
    v8f accs[2] = {acc0, acc1};
#pragma unroll
    for (int i = 0; i < 2; ++i) {
#pragma unroll
        for (int v = 0; v < 8; ++v) {
            const int m = 16 * i + v + 8 * half;
            const int node = n0 + m;
            if (node < N) {
                float val = accs[i][v] + bias;
                if (ACT) val = val > 0.0f ? val : 0.0f;
                out[(size_t)node * 128 + col] = val;
            }
        }
    }
}

// ---------------------------------------------------------------------------
// Host-side launch: 3 layers x (zero agg -> edge scatter -> node update).
// ---------------------------------------------------------------------------
extern "C" void kernel_launch(void* const* d_in, const int* in_sizes, int n_in,
                              void* d_out, int out_size, void* d_ws, size_t ws_size,
                              hipStream_t stream) {
    const float* nf    = (const float*)d_in[0];   // [N,64]
    const float* ef    = (const float*)d_in[1];   // [E,32]
    const int*   src   = (const int*)d_in[2];
    const int*   dst   = (const int*)d_in[3];
    const int*   etype = (const int*)d_in[4];
    const float* mw0 = (const float*)d_in[5];  const float* mb0 = (const float*)d_in[6];
    const float* nw0 = (const float*)d_in[7];  const float* nb0 = (const float*)d_in[8];
    const float* mw1 = (const float*)d_in[9];  const float* mb1 = (const float*)d_in[10];
    const float* nw1 = (const float*)d_in[11]; const float* nb1 = (const float*)d_in[12];
    const float* mw2 = (const float*)d_in[13]; const float* mb2 = (const float*)d_in[14];
    const float* nw2 = (const float*)d_in[15]; const float* nb2 = (const float*)d_in[16];

    const int N = in_sizes[0] / 64;
    const int E = in_sizes[2];

    float* bufA = (float*)d_ws;                 // [N,128]
    float* bufB = bufA + (size_t)N * 128;       // [N,128]
    float* agg  = bufB + (size_t)N * 128;       // [N,384]

    const int edgeBlocks = (E + 63) / 64;
    const int nodeBlocks = (N + 31) / 32;
    const long aggElems  = (long)N * 384;
    const int zeroBlocks = 1024;

    // ---- Layer 0: h_in = nf (DIN=64), no nf-concat, ReLU ----
    zero_f32<<<zeroBlocks, 256, 0, stream>>>(agg, aggElems);
    edge_msg_kernel<64><<<edgeBlocks, 256, 0, stream>>>(
        nf, ef, src, dst, etype, mw0, mb0, agg, E);
    node_update_kernel<64, false, true><<<nodeBlocks, 256, 0, stream>>>(
        nf, agg, nf, nw0, nb0, bufA, N);

    // ---- Layer 1: h_in = bufA (DIN=128), nf-concat, ReLU ----
    zero_f32<<<zeroBlocks, 256, 0, stream>>>(agg, aggElems);
    edge_msg_kernel<128><<<edgeBlocks, 256, 0, stream>>>(
        bufA, ef, src, dst, etype, mw1, mb1, agg, E);
    node_update_kernel<128, true, true><<<nodeBlocks, 256, 0, stream>>>(
        bufA, agg, nf, nw1, nb1, bufB, N);

    // ---- Layer 2: h_in = bufB (DIN=128), nf-concat, no ReLU ----
    zero_f32<<<zeroBlocks, 256, 0, stream>>>(agg, aggElems);
    edge_msg_kernel<128><<<edgeBlocks, 256, 0, stream>>>(
        bufB, ef, src, dst, etype, mw2, mb2, agg, E);
    node_update_kernel<128, true, false><<<nodeBlocks, 256, 0, stream>>>(
        bufB, agg, nf, nw2, nb2, (float*)d_out, N);
}